// MemoryEfficientMoE_22531398435276
// MI455X (gfx1250) — compile-verified
//
#include <hip/hip_runtime.h>
#include <math.h>

#define TOKENS 8192
#define DDIM   1024
#define NEXP   8
#define HDIM   4096
#define TOPK   2
#define LDSS   36   // LDS row stride (32 + 4 pad); 36*4=144 bytes = 9*16 keeps float4 alignment

typedef float v2f __attribute__((ext_vector_type(2)));
typedef float v8f __attribute__((ext_vector_type(8)));
typedef int   v4i __attribute__((vector_size(16)));   // matches builtin param pointee

#define LDS_AS __attribute__((address_space(3)))
#define GLB_AS __attribute__((address_space(1)))

#if __has_builtin(__builtin_amdgcn_global_load_async_to_lds_b128)
#define USE_ASYNC_BUILTIN 1
#else
#define USE_ASYNC_BUILTIN 0
#endif

// Async DMA: 16 bytes global -> LDS per lane, tracked by ASYNCcnt (no VGPR round trip)
__device__ __forceinline__ void async_copy_b128(const float* gsrc, float* ldst) {
#if USE_ASYNC_BUILTIN
  __builtin_amdgcn_global_load_async_to_lds_b128(
      (GLB_AS v4i*)gsrc, (LDS_AS v4i*)ldst, 0, 0);
#else
  const unsigned lofs = (unsigned)(unsigned long long)(LDS_AS void*)ldst;
  asm volatile("global_load_async_to_lds_b128 %0, %1, off"
               :: "v"(lofs), "v"(gsrc) : "memory");
#endif
}

__device__ __forceinline__ void wait_async0() {
#if __has_builtin(__builtin_amdgcn_s_wait_asynccnt)
  __builtin_amdgcn_s_wait_asynccnt(0);
#else
  asm volatile("s_wait_asynccnt 0x0" ::: "memory");
#endif
}

__device__ __forceinline__ float gelu_exact(float x) {
  return 0.5f * x * (1.0f + erff(x * 0.70710678118654752440f));
}

// -------------------- router: LN -> logits -> softmax top2 --------------------
__global__ __launch_bounds__(256) void router_kernel(
    const float* __restrict__ x, const float* __restrict__ rls,
    const float* __restrict__ rlb, const float* __restrict__ rw,
    const float* __restrict__ rbias, float* __restrict__ xhat,
    int* __restrict__ topi, float* __restrict__ topw, int* __restrict__ cnt) {
  const int t   = blockIdx.x;
  const int tid = threadIdx.x;
  const int d0  = tid * 4;

  __shared__ float s1[256];
  __shared__ float s2[256];
  __shared__ float sl[8 * 256];

  const float4 xv = *(const float4*)(x + (size_t)t * DDIM + d0);
  s1[tid] = xv.x + xv.y + xv.z + xv.w;
  s2[tid] = xv.x*xv.x + xv.y*xv.y + xv.z*xv.z + xv.w*xv.w;
  __syncthreads();
  for (int off = 128; off > 0; off >>= 1) {
    if (tid < off) { s1[tid] += s1[tid + off]; s2[tid] += s2[tid + off]; }
    __syncthreads();
  }
  const float mean = s1[0] * (1.0f / DDIM);
  const float var  = s2[0] * (1.0f / DDIM) - mean * mean;
  const float rstd = rsqrtf(var + 1e-6f);

  float4 hv;
  hv.x = (xv.x - mean) * rstd;
  hv.y = (xv.y - mean) * rstd;
  hv.z = (xv.z - mean) * rstd;
  hv.w = (xv.w - mean) * rstd;
  *(float4*)(xhat + (size_t)t * DDIM + d0) = hv;   // shared x-hat (expert LN reuses it)

  const float4 sv = *(const float4*)(rls + d0);
  const float4 bv = *(const float4*)(rlb + d0);
  const float r0 = hv.x * sv.x + bv.x;
  const float r1 = hv.y * sv.y + bv.y;
  const float r2 = hv.z * sv.z + bv.z;
  const float r3 = hv.w * sv.w + bv.w;

#pragma unroll
  for (int e = 0; e < NEXP; ++e) {
    const float4 wv = *(const float4*)(rw + (size_t)e * DDIM + d0);
    sl[e * 256 + tid] = r0 * wv.x + r1 * wv.y + r2 * wv.z + r3 * wv.w;
  }
  __syncthreads();
  for (int off = 128; off > 0; off >>= 1) {
    if (tid < off) {
#pragma unroll
      for (int e = 0; e < NEXP; ++e) sl[e * 256 + tid] += sl[e * 256 + tid + off];
    }
    __syncthreads();
  }

  if (tid == 0) {
    float lg[NEXP];
#pragma unroll
    for (int e = 0; e < NEXP; ++e) lg[e] = sl[e * 256] + rbias[e];
    int i0 = 0;
#pragma unroll
    for (int e = 1; e < NEXP; ++e) if (lg[e] > lg[i0]) i0 = e;
    int i1 = (i0 == 0) ? 1 : 0;
#pragma unroll
    for (int e = 0; e < NEXP; ++e) if (e != i0 && lg[e] > lg[i1]) i1 = e;
    const float e1 = expf(lg[i1] - lg[i0]);
    const float w0 = 1.0f / (1.0f + e1);
    const float w1 = e1 * w0;
    topi[t * 2 + 0] = i0;  topw[t * 2 + 0] = w0;
    topi[t * 2 + 1] = i1;  topw[t * 2 + 1] = w1;
    atomicAdd(&cnt[i0], 1);
    atomicAdd(&cnt[i1], 1);
  }
}

// -------------------- offsets (prefix sum over 8 experts) --------------------
__global__ void offsets_kernel(const int* __restrict__ cnt, int* __restrict__ off,
                               int* __restrict__ cursor) {
  if (threadIdx.x == 0 && blockIdx.x == 0) {
    int a = 0;
    for (int e = 0; e < NEXP; ++e) { off[e] = a; a += cnt[e]; cursor[e] = 0; }
    off[NEXP] = a;
  }
}

// -------------------- scatter: compact per-expert token lists --------------------
__global__ __launch_bounds__(256) void scatter_kernel(
    const int* __restrict__ topi, const float* __restrict__ topw,
    const int* __restrict__ off, int* __restrict__ cursor,
    int* __restrict__ tok_list, float* __restrict__ gate_list,
    int* __restrict__ slot_of) {
  const int t = blockIdx.x * 256 + threadIdx.x;
  if (t >= TOKENS) return;
#pragma unroll
  for (int k = 0; k < TOPK; ++k) {
    const int e  = topi[t * 2 + k];
    const int sl = atomicAdd(&cursor[e], 1);
    const int pos = off[e] + sl;
    tok_list[pos]  = t;
    gate_list[pos] = topw[t * 2 + k];
    slot_of[t * 2 + k] = pos;
  }
}

// ------- tiled WMMA GEMM, double-buffered async LDS staging -------------------
// MODE 1: h = GELU((xhat*sE+bE) @ W1^T + b1)    MODE 2: y = gate*(h @ W2^T + b2)
template <int MODE>
__global__ __launch_bounds__(256) void gemm_kernel(
    const float* __restrict__ Ax,   // xhat [T, D]      (MODE 1)
    const float* __restrict__ Ah,   // h    [slots, H]  (MODE 2)
    const float* __restrict__ W,    // w1 [E,H,D] or w2 [E,D,H]
    const float* __restrict__ bias, // b1 [E,H]  or b2 [E,D]
    const float* __restrict__ els,  // expert_ln_scale [E,D] (MODE 1)
    const float* __restrict__ elb,  // expert_ln_bias  [E,D] (MODE 1)
    const int* __restrict__ off, const int* __restrict__ cnt,
    const int* __restrict__ tok_list, const float* __restrict__ gate_list,
    float* __restrict__ Obuf)       // h [slots,H] or y [slots,D]
{
  constexpr int KD = (MODE == 1) ? DDIM : HDIM;
  constexpr int ND = (MODE == 1) ? HDIM : DDIM;

  const int e    = blockIdx.z;
  const int n0   = blockIdx.x * 128;
  const int m0   = blockIdx.y * 128;
  const int cntE = cnt[e];
  if (m0 >= cntE) return;        // block-uniform early exit
  const int offE = off[e];

  __shared__ float sA[2][128 * LDSS];
  __shared__ float sB[2][128 * LDSS];
  __shared__ int   sTok[128];
  __shared__ float sGate[128];

  const int tid = threadIdx.x;
  if (tid < 128) {
    const int s = m0 + tid;
    if (s < cntE) { sTok[tid] = tok_list[offE + s]; sGate[tid] = gate_list[offE + s]; }
    else          { sTok[tid] = -1;                 sGate[tid] = 0.0f; }
  }
  __syncthreads();

  const int wid  = tid >> 5;
  const int lane = tid & 31;
  const int wm   = wid & 3;     // 4 row-groups of 32 rows
  const int wn   = wid >> 2;    // 2 col-groups of 64 cols
  const int lm   = lane & 15;
  const int kh   = lane >> 4;   // K-half selector for f32 WMMA A/B frags

  // staging geometry: thread covers 4 rows (rr[i]) at a fixed 4-col span cA
  const int cA = (tid & 7) << 2;
  int rr[4];
#pragma unroll
  for (int i = 0; i < 4; ++i) rr[i] = (tid + i * 256) >> 3;

  float4 pxv[4], psv, pbv;     // MODE-1 A pipeline registers

  auto issueB = [&](int k0, int buf) {   // weights: async DMA to LDS
#pragma unroll
    for (int i = 0; i < 4; ++i)
      async_copy_b128(W + ((size_t)e * ND + (n0 + rr[i])) * KD + k0 + cA,
                      &sB[buf][rr[i] * LDSS + cA]);
  };
  auto issueA2 = [&](int k0, int buf) {  // MODE 2 A (h rows): async DMA, clamp OOR rows
#pragma unroll
    for (int i = 0; i < 4; ++i) {
      const int s  = m0 + rr[i];
      const int sc = (s < cntE) ? s : 0;   // masked in epilogue; keep address in-bounds
      async_copy_b128(Ah + (size_t)(offE + sc) * HDIM + k0 + cA,
                      &sA[buf][rr[i] * LDSS + cA]);
    }
  };
  auto loadA1 = [&](int k0) {            // MODE 1 A: global loads into regs
    psv = *(const float4*)(els + (size_t)e * DDIM + k0 + cA);
    pbv = *(const float4*)(elb + (size_t)e * DDIM + k0 + cA);
#pragma unroll
    for (int i = 0; i < 4; ++i) {
      int tok = sTok[rr[i]];
      if (tok < 0) tok = 0;              // masked in epilogue; keep address in-bounds
      pxv[i] = *(const float4*)(Ax + (size_t)tok * DDIM + k0 + cA);
    }
  };
  auto storeA1 = [&](int buf) {          // MODE 1 A: apply expert LN affine, store to LDS
#pragma unroll
    for (int i = 0; i < 4; ++i) {
      float4 av;
      av.x = pxv[i].x * psv.x + pbv.x;
      av.y = pxv[i].y * psv.y + pbv.y;
      av.z = pxv[i].z * psv.z + pbv.z;
      av.w = pxv[i].w * psv.w + pbv.w;
      *(float4*)(&sA[buf][rr[i] * LDSS + cA]) = av;
    }
  };

  v8f acc[2][4];
#pragma unroll
  for (int i = 0; i < 2; ++i)
#pragma unroll
    for (int j = 0; j < 4; ++j)
      acc[i][j] = v8f{0.f, 0.f, 0.f, 0.f, 0.f, 0.f, 0.f, 0.f};

  // prologue: stage k0 = 0 into buffer 0
  if (MODE == 1) loadA1(0); else issueA2(0, 0);
  issueB(0, 0);
  if (MODE == 1) storeA1(0);
  wait_async0();
  __syncthreads();

  int cur = 0;
  for (int k0 = 0; k0 < KD; k0 += 32) {
    const bool more = (k0 + 32) < KD;
    const int  nxt  = cur ^ 1;
    if (more) {                          // prefetch next stage while computing
      if (MODE == 1) loadA1(k0 + 32); else issueA2(k0 + 32, nxt);
      issueB(k0 + 32, nxt);
    }

    // 8 k-steps of 4; 2x4 WMMA tiles per wave: v_wmma_f32_16x16x4_f32
#pragma unroll
    for (int kk = 0; kk < 32; kk += 4) {
      v2f aF[2], bF[4];
#pragma unroll
      for (int tm = 0; tm < 2; ++tm)
        aF[tm] = *reinterpret_cast<const v2f*>(
            &sA[cur][(wm * 32 + tm * 16 + lm) * LDSS + kk + 2 * kh]);
#pragma unroll
      for (int tn = 0; tn < 4; ++tn)
        bF[tn] = *reinterpret_cast<const v2f*>(
            &sB[cur][(wn * 64 + tn * 16 + lm) * LDSS + kk + 2 * kh]);
#pragma unroll
      for (int tm = 0; tm < 2; ++tm)
#pragma unroll
        for (int tn = 0; tn < 4; ++tn)
          acc[tm][tn] = __builtin_amdgcn_wmma_f32_16x16x4_f32(
              false, aF[tm], false, bF[tn], (short)0, acc[tm][tn], false, false);
    }

    if (more && MODE == 1) storeA1(nxt); // global loads landed during compute
    if (more) wait_async0();             // my async copies into nxt are done
    __syncthreads();                     // everyone's copies visible; cur fully consumed
    cur = nxt;
  }

  // epilogue; C/D layout: VGPR v -> M = v + 8*(lane>=16), N = lane&15
#pragma unroll
  for (int tm = 0; tm < 2; ++tm) {
#pragma unroll
    for (int tn = 0; tn < 4; ++tn) {
      const int nG = n0 + wn * 64 + tn * 16 + lm;
#pragma unroll
      for (int v = 0; v < 8; ++v) {
        const int m = wm * 32 + tm * 16 + v + 8 * kh;
        const int s = m0 + m;
        if (s < cntE) {
          float val = acc[tm][tn][v] + bias[(size_t)e * ND + nG];
          if (MODE == 1) {
            Obuf[(size_t)(offE + s) * ND + nG] = gelu_exact(val);
          } else {
            Obuf[(size_t)(offE + s) * ND + nG] = val * sGate[m];
          }
        }
      }
    }
  }
}

// -------------------- combine: out[t] = y[slot(t,0)] + y[slot(t,1)] ------------
__global__ __launch_bounds__(256) void combine_kernel(
    const float* __restrict__ y, const int* __restrict__ slot_of,
    float* __restrict__ out) {
  const int gid  = blockIdx.x * 256 + threadIdx.x;
  const int base = gid * 4;
  const int t    = base / DDIM;
  const int d    = base - t * DDIM;
  const int p0 = slot_of[t * 2 + 0];
  const int p1 = slot_of[t * 2 + 1];
  const float4 a = *(const float4*)(y + (size_t)p0 * DDIM + d);
  const float4 b = *(const float4*)(y + (size_t)p1 * DDIM + d);
  float4 o;
  o.x = a.x + b.x;  o.y = a.y + b.y;  o.z = a.z + b.z;  o.w = a.w + b.w;
  *(float4*)(out + base) = o;
}

// -------------------- launch --------------------
extern "C" void kernel_launch(void* const* d_in, const int* in_sizes, int n_in,
                              void* d_out, int out_size, void* d_ws, size_t ws_size,
                              hipStream_t stream) {
  const float* x    = (const float*)d_in[0];
  const float* rls  = (const float*)d_in[1];
  const float* rlb  = (const float*)d_in[2];
  const float* rw   = (const float*)d_in[3];
  const float* rbb  = (const float*)d_in[4];
  const float* els  = (const float*)d_in[5];
  const float* elb  = (const float*)d_in[6];
  const float* w1   = (const float*)d_in[7];
  const float* b1   = (const float*)d_in[8];
  const float* w2   = (const float*)d_in[9];
  const float* b2   = (const float*)d_in[10];
  float* out = (float*)d_out;

  char* p = (char*)d_ws;
  auto wsalloc = [&](size_t bytes) -> void* {
    void* r = (void*)p;
    p += (bytes + 255) & ~(size_t)255;
    return r;
  };
  float* xhat      = (float*)wsalloc((size_t)TOKENS * DDIM * 4);          //  33.5 MB
  float* hbuf      = (float*)wsalloc((size_t)TOKENS * TOPK * HDIM * 4);   // 268.4 MB
  float* ybuf      = (float*)wsalloc((size_t)TOKENS * TOPK * DDIM * 4);   //  67.1 MB
  float* topw      = (float*)wsalloc((size_t)TOKENS * TOPK * 4);
  float* gate_list = (float*)wsalloc((size_t)TOKENS * TOPK * 4);
  int*   topi      = (int*)wsalloc((size_t)TOKENS * TOPK * 4);
  int*   tok_list  = (int*)wsalloc((size_t)TOKENS * TOPK * 4);
  int*   slot_of   = (int*)wsalloc((size_t)TOKENS * TOPK * 4);
  int*   cnt       = (int*)wsalloc(256);
  int*   off       = (int*)wsalloc(256);
  int*   cursor    = (int*)wsalloc(256);

  (void)hipMemsetAsync(cnt, 0, NEXP * sizeof(int), stream);

  router_kernel<<<TOKENS, 256, 0, stream>>>(x, rls, rlb, rw, rbb, xhat, topi, topw, cnt);
  offsets_kernel<<<1, 64, 0, stream>>>(cnt, off, cursor);
  scatter_kernel<<<TOKENS / 256, 256, 0, stream>>>(topi, topw, off, cursor,
                                                   tok_list, gate_list, slot_of);

  dim3 g1(HDIM / 128, TOKENS / 128, NEXP);
  gemm_kernel<1><<<g1, 256, 0, stream>>>(xhat, nullptr, w1, b1, els, elb,
                                         off, cnt, tok_list, gate_list, hbuf);

  dim3 g2(DDIM / 128, TOKENS / 128, NEXP);
  gemm_kernel<2><<<g2, 256, 0, stream>>>(nullptr, hbuf, w2, b2, nullptr, nullptr,
                                         off, cnt, tok_list, gate_list, ybuf);

  combine_kernel<<<(TOKENS * DDIM / 4) / 256, 256, 0, stream>>>(ybuf, slot_of, out);
}